// concat_attention_23854248362442
// MI455X (gfx1250) — compile-verified
//
#include <hip/hip_runtime.h>
#include <hip/hip_bf16.h>

// Problem sizes (fixed by the reference): B=4, Sx=Sy=512, H=128, D=2*H=256.
#define BB 4
#define SS 512
#define HH 128
#define DD 256

typedef __attribute__((ext_vector_type(2))) float v2f;
typedef __attribute__((ext_vector_type(8))) float v8f;
typedef __attribute__((ext_vector_type(4))) int   v4i;

#define L2E 1.4426950408889634f

#define AS1 __attribute__((address_space(1)))
#define AS3 __attribute__((address_space(3)))

#if __has_builtin(__builtin_amdgcn_global_load_async_to_lds_b128) && \
    __has_builtin(__builtin_amdgcn_s_wait_asynccnt)
#define HAVE_ASYNC_LDS 1
#else
#define HAVE_ASYNC_LDS 0
#endif

__device__ __forceinline__ float fast_tanh(float x) {
#if __has_builtin(__builtin_amdgcn_tanhf)
  return __builtin_amdgcn_tanhf(x);          // gfx1250 v_tanh_f32 (TRANS)
#else
  // tanh(x) = 1 - 2/(exp(2x)+1), via hardware v_exp_f32 / v_rcp_f32
  float e = __builtin_amdgcn_exp2f(x * (2.0f * L2E));
  return 1.0f - 2.0f * __builtin_amdgcn_rcpf(e + 1.0f);
#endif
}

// ---------------------------------------------------------------------------
// Pass 1: S = X @ W^T via V_WMMA_F32_16X16X4_F32.
// X: (2048, 256) row-major, W: (128, 256) row-major, S: (2048, 128).
// One wave (32 lanes) computes one 16x16 tile of S, K-loop over 256 in steps
// of 4.  A-matrix layout (ISA 7.12.2, 32-bit A 16x4): lane m=l&15,
// lanes 0-15 hold K=k+0 (v0), k+1 (v1); lanes 16-31 hold K=k+2, k+3.
// B-matrix (4x16) mirrors this with N = l&15.
// ---------------------------------------------------------------------------
__global__ void proj_wmma_kernel(const float* __restrict__ X,
                                 const float* __restrict__ W,
                                 float* __restrict__ S) {
  const int tile = blockIdx.x;          // 128 M-tiles x 8 N-tiles = 1024
  const int m0 = (tile >> 3) * 16;
  const int n0 = (tile & 7) * 16;
  const int lane = threadIdx.x;         // wave32
  const int mn   = lane & 15;
  const int koff = (lane >> 4) * 2;

  const float* arow = X + (size_t)(m0 + mn) * DD + koff;  // A: X rows (contig K)
  const float* brow = W + (size_t)(n0 + mn) * DD + koff;  // B[k][n] = W[n][k]
  v8f c = {};
  for (int k = 0; k < DD; k += 4) {
    v2f a = *(const v2f*)(arow + k);
    v2f b = *(const v2f*)(brow + k);
    c = __builtin_amdgcn_wmma_f32_16x16x4_f32(false, a, false, b,
                                              (short)0, c, false, false);
  }
  const int mhi = (lane >> 4) * 8;      // C/D: VGPR j -> M = j + 8*(lane>=16)
#pragma unroll
  for (int j = 0; j < 8; ++j)
    S[(size_t)(m0 + mhi + j) * HH + n0 + mn] = c[j];
}

// ---------------------------------------------------------------------------
// Pass 2: scores sj[b,t,s] = sum_h tanh(s1[b,s,h] + s2[b,t,h]) * vc[h].
// 16x16 (t,s) tile per 256-thread block; s1/s2 tiles + vc staged in LDS with
// padded stride (132 floats) for conflict-free b128 reads.  Staging uses the
// CDNA5 async global->LDS path (ASYNCcnt) when available.  This is the hot
// loop: 134M tanh total, TRANS-bound.
// ---------------------------------------------------------------------------
__global__ void score_tanh_kernel(const float* __restrict__ S1,
                                  const float* __restrict__ S2,
                                  const float* __restrict__ vc,
                                  float* __restrict__ SJ) {
  __shared__ float s1t[16 * 132];
  __shared__ float s2t[16 * 132];
  __shared__ float vcs[HH];

  const int b  = blockIdx.z;
  const int s0 = blockIdx.x * 16;
  const int t0 = blockIdx.y * 16;
  const int tid = threadIdx.x;          // 256

  // Stage tiles: 16 rows x 128 cols = 512 float4 per tile, 2 per thread.
#if HAVE_ASYNC_LDS
  {
    int idx = tid;
#pragma unroll
    for (int it = 0; it < 2; ++it, idx += 256) {
      const int row = idx >> 5;           // 0..15
      const int c4  = (idx & 31) * 4;     // 0..124
      __builtin_amdgcn_global_load_async_to_lds_b128(
          (AS1 v4i*)(S1 + (size_t)((b * SS) + s0 + row) * HH + c4),
          (AS3 v4i*)(s1t + row * 132 + c4), 0, 0);
      __builtin_amdgcn_global_load_async_to_lds_b128(
          (AS1 v4i*)(S2 + (size_t)((b * SS) + t0 + row) * HH + c4),
          (AS3 v4i*)(s2t + row * 132 + c4), 0, 0);
    }
    if (tid < 32)                         // 32 lanes x 16B = 512B = vc[128]
      __builtin_amdgcn_global_load_async_to_lds_b128(
          (AS1 v4i*)(vc + tid * 4), (AS3 v4i*)(vcs + tid * 4), 0, 0);
    __builtin_amdgcn_s_wait_asynccnt(0);
  }
#else
  {
    int idx = tid;
#pragma unroll
    for (int it = 0; it < 2; ++it, idx += 256) {
      const int row = idx >> 5;           // 0..15
      const int c4  = (idx & 31) * 4;     // 0..124
      float4 a = *(const float4*)(S1 + (size_t)((b * SS) + s0 + row) * HH + c4);
      float4 bq = *(const float4*)(S2 + (size_t)((b * SS) + t0 + row) * HH + c4);
      *(float4*)(s1t + row * 132 + c4) = a;
      *(float4*)(s2t + row * 132 + c4) = bq;
    }
    if (tid < HH) vcs[tid] = vc[tid];
  }
#endif
  __syncthreads();

  const int sl = tid & 15;
  const int tl = tid >> 4;
  const float* r1 = s1t + sl * 132;
  const float* r2 = s2t + tl * 132;

  float acc = 0.0f;
#pragma unroll 8
  for (int h = 0; h < HH; h += 4) {
    float4 a = *(const float4*)(r1 + h);
    float4 bq = *(const float4*)(r2 + h);
    float4 v = *(const float4*)(vcs + h);
    acc += fast_tanh(a.x + bq.x) * v.x;
    acc += fast_tanh(a.y + bq.y) * v.y;
    acc += fast_tanh(a.z + bq.z) * v.z;
    acc += fast_tanh(a.w + bq.w) * v.w;
  }
  SJ[(size_t)((b * SS) + t0 + tl) * SS + s0 + sl] = acc;
}

// ---------------------------------------------------------------------------
// Pass 3: in-place row softmax over Sx (512 elems), one block per (b,t) row.
// ---------------------------------------------------------------------------
__global__ void softmax_row_kernel(float* __restrict__ SJ) {
  __shared__ float red[256];
  float* p = SJ + (size_t)blockIdx.x * SS;
  const int tid = threadIdx.x;

  float v0 = p[tid];
  float v1 = p[tid + 256];
  red[tid] = fmaxf(v0, v1);
  __syncthreads();
  for (int s = 128; s > 0; s >>= 1) {
    if (tid < s) red[tid] = fmaxf(red[tid], red[tid + s]);
    __syncthreads();
  }
  const float M = red[0];
  __syncthreads();
  float e0 = __builtin_amdgcn_exp2f((v0 - M) * L2E);
  float e1 = __builtin_amdgcn_exp2f((v1 - M) * L2E);
  red[tid] = e0 + e1;
  __syncthreads();
  for (int s = 128; s > 0; s >>= 1) {
    if (tid < s) red[tid] += red[tid + s];
    __syncthreads();
  }
  const float inv = 1.0f / red[0];
  p[tid]       = e0 * inv;
  p[tid + 256] = e1 * inv;
}

// ---------------------------------------------------------------------------
// Pass 4: qtc = ait @ x via V_WMMA_F32_16X16X4_F32.
// A: (B,512,512) attn weights, X: (B,512,256), O: (B,512,256).
// One wave per 16x16 output tile, K over 512 in steps of 4.
// ---------------------------------------------------------------------------
__global__ void qtc_wmma_kernel(const float* __restrict__ A,
                                const float* __restrict__ X,
                                float* __restrict__ O) {
  const int b  = blockIdx.z;
  const int m0 = blockIdx.y * 16;
  const int n0 = blockIdx.x * 16;
  const int lane = threadIdx.x;
  const int mn   = lane & 15;
  const int koff = (lane >> 4) * 2;

  const float* arow = A + (size_t)((b * SS) + m0 + mn) * SS + koff;
  const float* xcol = X + (size_t)b * SS * DD + n0 + mn;
  v8f c = {};
  for (int k = 0; k < SS; k += 4) {
    v2f a = *(const v2f*)(arow + k);
    v2f bb;
    bb.x = xcol[(size_t)(k + koff) * DD];       // B[k][n] = x[b, k, n]
    bb.y = xcol[(size_t)(k + koff + 1) * DD];
    c = __builtin_amdgcn_wmma_f32_16x16x4_f32(false, a, false, bb,
                                              (short)0, c, false, false);
  }
  const int mhi = (lane >> 4) * 8;
#pragma unroll
  for (int j = 0; j < 8; ++j)
    O[(size_t)((b * SS) + m0 + mhi + j) * DD + n0 + mn] = c[j];
}

// ---------------------------------------------------------------------------
// Workspace layout (floats): s1[2048*128] | s2[2048*128] | sj[4*512*512]
// = 1 MB + 1 MB + 4 MB = 6 MB.
// ---------------------------------------------------------------------------
extern "C" void kernel_launch(void* const* d_in, const int* in_sizes, int n_in,
                              void* d_out, int out_size, void* d_ws, size_t ws_size,
                              hipStream_t stream) {
  (void)in_sizes; (void)n_in; (void)out_size; (void)ws_size;
  const float* x   = (const float*)d_in[0];  // (B, Sx, 2H)
  const float* y   = (const float*)d_in[1];  // (B, Sy, 2H)
  const float* Wc1 = (const float*)d_in[2];  // (H, 2H)
  const float* Wc2 = (const float*)d_in[3];  // (H, 2H)
  const float* vc  = (const float*)d_in[4];  // (1, H)
  float* out = (float*)d_out;                // (B, Sy, 2H)

  float* ws = (float*)d_ws;
  float* s1 = ws;                            // (B*Sx, H)
  float* s2 = s1 + (size_t)BB * SS * HH;     // (B*Sy, H)
  float* sj = s2 + (size_t)BB * SS * HH;     // (B, Sy, Sx)

  // Projections (f32 WMMA GEMMs)
  proj_wmma_kernel<<<dim3(1024), dim3(32), 0, stream>>>(x, Wc1, s1);
  proj_wmma_kernel<<<dim3(1024), dim3(32), 0, stream>>>(y, Wc2, s2);
  // Additive-tanh scores (dominant cost, v_tanh_f32)
  score_tanh_kernel<<<dim3(SS / 16, SS / 16, BB), dim3(256), 0, stream>>>(s1, s2, vc, sj);
  // Row softmax over Sx
  softmax_row_kernel<<<dim3(BB * SS), dim3(256), 0, stream>>>(sj);
  // Weighted sum qtc = ait @ x (f32 WMMA GEMM)
  qtc_wmma_kernel<<<dim3(DD / 16, SS / 16, BB), dim3(32), 0, stream>>>(sj, x, out);
}